// ModernHopfieldNetwork_77816217469499
// MI455X (gfx1250) — compile-verified
//
#include <hip/hip_runtime.h>
#include <hip/hip_bf16.h>

typedef float v8f __attribute__((ext_vector_type(8)));
typedef float v2f __attribute__((ext_vector_type(2)));

#define B_      2048
#define N_      50000
#define D_      128
#define NSPLIT  16
#define NCHUNK  (N_ / NSPLIT)     // 3125 (exact)
#define BT      16
#define NBT     (B_ / BT)         // 128
#define NREC    (NBT * NSPLIT)    // 2048 partial records
#define BETA_   100.0f
#define SMASK   (-1.0e11f)        // BETA * (-1e9) masked fill, in s-units
#define SPAD    (-3.0e38f)        // padding sentinel (exp -> 0, stays finite)

// workspace layout (floats):
//   zp  [NREC][16][128], mp/lp/bvp [NREC][16], bip [NREC][16] (int)   ~17.3 MB

// One 16-memory group: L1 distances (half-D per wave half, shfl-combined),
// online softmax update, then P @ M_tile via 32x V_WMMA_F32_16X16X4_F32.
// TAIL=true adds bounds clamping (only used for the final 5-valid group).
template <bool TAIL>
__device__ __forceinline__ void process_group(
    int nb, int n_end, int r, int h,
    const float* __restrict__ M, const unsigned char* __restrict__ mask,
    const float (&qv)[64], v8f (&C)[8],
    float& Ms, float& L, float& bv, int& bi)
{
    float S[16];

    // ---- L1 distances: lane (r,h) accumulates d in [h*64, h*64+64) for 4 rows/sg ----
    const float4* base4 = (const float4*)(M + (size_t)nb * D_ + h * 64); // fast path base
#pragma unroll
    for (int sg = 0; sg < 4; ++sg) {
        const float4* rp0;
        const float4* rp1;
        const float4* rp2;
        const float4* rp3;
        if constexpr (TAIL) {
            int n0c = nb + 4 * sg + 0; n0c = (n0c < n_end) ? n0c : (n_end - 1);
            int n1c = nb + 4 * sg + 1; n1c = (n1c < n_end) ? n1c : (n_end - 1);
            int n2c = nb + 4 * sg + 2; n2c = (n2c < n_end) ? n2c : (n_end - 1);
            int n3c = nb + 4 * sg + 3; n3c = (n3c < n_end) ? n3c : (n_end - 1);
            rp0 = (const float4*)(M + (size_t)n0c * D_ + h * 64);
            rp1 = (const float4*)(M + (size_t)n1c * D_ + h * 64);
            rp2 = (const float4*)(M + (size_t)n2c * D_ + h * 64);
            rp3 = (const float4*)(M + (size_t)n3c * D_ + h * 64);
        } else {
            rp0 = base4 + (4 * sg + 0) * (D_ / 4);   // immediate offsets from one base
            rp1 = base4 + (4 * sg + 1) * (D_ / 4);
            rp2 = base4 + (4 * sg + 2) * (D_ / 4);
            rp3 = base4 + (4 * sg + 3) * (D_ / 4);
        }
        float d0 = 0.f, d1 = 0.f, d2 = 0.f, d3 = 0.f;
#pragma unroll
        for (int i = 0; i < 16; ++i) {
            float4 m0 = rp0[i];
            float4 m1 = rp1[i];
            float4 m2 = rp2[i];
            float4 m3 = rp3[i];
            d0 += fabsf(qv[4*i+0]-m0.x)+fabsf(qv[4*i+1]-m0.y)+fabsf(qv[4*i+2]-m0.z)+fabsf(qv[4*i+3]-m0.w);
            d1 += fabsf(qv[4*i+0]-m1.x)+fabsf(qv[4*i+1]-m1.y)+fabsf(qv[4*i+2]-m1.z)+fabsf(qv[4*i+3]-m1.w);
            d2 += fabsf(qv[4*i+0]-m2.x)+fabsf(qv[4*i+1]-m2.y)+fabsf(qv[4*i+2]-m2.z)+fabsf(qv[4*i+3]-m2.w);
            d3 += fabsf(qv[4*i+0]-m3.x)+fabsf(qv[4*i+1]-m3.y)+fabsf(qv[4*i+2]-m3.z)+fabsf(qv[4*i+3]-m3.w);
        }
        // combine the two d-halves: every lane gets the full distance of all 4 rows
        d0 += __shfl_xor(d0, 16, 32);
        d1 += __shfl_xor(d1, 16, 32);
        d2 += __shfl_xor(d2, 16, 32);
        d3 += __shfl_xor(d3, 16, 32);
        const int n0 = nb + 4 * sg;
        bool v0 = !TAIL || (n0 + 0 < n_end);
        bool v1 = !TAIL || (n0 + 1 < n_end);
        bool v2 = !TAIL || (n0 + 2 < n_end);
        bool v3 = !TAIL || (n0 + 3 < n_end);
        S[4*sg+0] = v0 ? (mask[n0 + 0] ? -BETA_ * d0 : SMASK) : SPAD;
        S[4*sg+1] = v1 ? (mask[n0 + 1] ? -BETA_ * d1 : SMASK) : SPAD;
        S[4*sg+2] = v2 ? (mask[n0 + 2] ? -BETA_ * d2 : SMASK) : SPAD;
        S[4*sg+3] = v3 ? (mask[n0 + 3] ? -BETA_ * d3 : SMASK) : SPAD;
    }

    // ---- online softmax update (every lane holds all 16 s-values) ----
    float gm = S[0];
#pragma unroll
    for (int k = 1; k < 16; ++k) gm = fmaxf(gm, S[k]);
    const float Mn    = fmaxf(Ms, gm);
    const float scale = __expf(Ms - Mn);
#pragma unroll
    for (int t = 0; t < 8; ++t)
#pragma unroll
        for (int k = 0; k < 8; ++k) C[t][k] *= scale;

    // select this lane's A-layout elements (h picks odd/even n), exp, argmax
    float Pa[4], Pb[4];
    float ps = 0.f;
#pragma unroll
    for (int sg = 0; sg < 4; ++sg) {
        const float sa = h ? S[4*sg+1] : S[4*sg+0];
        const float sb = h ? S[4*sg+3] : S[4*sg+2];
        const int n_a = nb + 4 * sg + h;
        const int n_b = nb + 4 * sg + 2 + h;
        if (sa > bv) { bv = sa; bi = n_a; }    // strict >: first occurrence wins
        if (sb > bv) { bv = sb; bi = n_b; }
        const float pa = __expf(sa - Mn);
        const float pb = __expf(sb - Mn);
        Pa[sg] = pa; Pb[sg] = pb;
        ps += pa + pb;
    }
    ps += __shfl_xor(ps, 16, 32);              // halves cover disjoint n
    L  = L * scale + ps;
    Ms = Mn;

    // ---- z += P @ M_tile via V_WMMA_F32_16X16X4_F32 ----
    // A: VGPR0=[K0|K2]=Pa, VGPR1=[K1|K3]=Pb (K0=n0, K2=n0+1, K1=n0+2, K3=n0+3)
    // B: VGPR0 = M[n0+h][col r], VGPR1 = M[n0+2+h][col r]
    const float* pbase = M + (size_t)(nb + h) * D_ + r;  // one base, immediate offsets
#pragma unroll
    for (int sg = 0; sg < 4; ++sg) {
        const float* Ba;
        const float* Bb;
        if constexpr (TAIL) {
            int na = nb + 4 * sg + h;     na = (na < n_end) ? na : (n_end - 1);
            int nbv = nb + 4 * sg + 2 + h; nbv = (nbv < n_end) ? nbv : (n_end - 1);
            Ba = M + (size_t)na  * D_ + r;
            Bb = M + (size_t)nbv * D_ + r;
        } else {
            Ba = pbase + (4 * sg) * D_;
            Bb = pbase + (4 * sg + 2) * D_;
        }
        v2f A;
        A.x = Pa[sg];
        A.y = Pb[sg];
#pragma unroll
        for (int t = 0; t < 8; ++t) {
            v2f Bv;
            Bv.x = Ba[t * 16];
            Bv.y = Bb[t * 16];
            C[t] = __builtin_amdgcn_wmma_f32_16x16x4_f32(
                false, A, false, Bv, (short)0, C[t], false, false);
        }
    }
}

__global__ __launch_bounds__(32, 4)
void hopfield_pass1(const float* __restrict__ q,
                    const float* __restrict__ M,
                    const unsigned char* __restrict__ mask,
                    float* __restrict__ ws) {
    const int lane  = threadIdx.x;
    const int r     = lane & 15;          // query row within tile / B column
    const int h     = lane >> 4;          // wave half = d-half owner
    const int btile = blockIdx.x;
    const int chunk = blockIdx.y;
    const int rec   = btile * NSPLIT + chunk;
    const int n_start = chunk * NCHUNK;
    const int n_end   = n_start + NCHUNK;

    // ---- half of q row in VGPRs: lane (r,h) holds q[row][h*64 .. h*64+63] ----
    float qv[64];
    {
        const float4* q4 = (const float4*)(q + (size_t)(btile * BT + r) * D_ + h * 64);
#pragma unroll
        for (int i = 0; i < 16; ++i) {
            float4 t = q4[i];
            qv[4*i+0] = t.x; qv[4*i+1] = t.y; qv[4*i+2] = t.z; qv[4*i+3] = t.w;
        }
    }

    v8f C[8];
#pragma unroll
    for (int t = 0; t < 8; ++t)
#pragma unroll
        for (int k = 0; k < 8; ++k) C[t][k] = 0.0f;

    float Ms = SPAD, L = 0.0f, bv = SPAD;
    int   bi = 0;

    const int NG = (NCHUNK + 15) / 16;    // 196; groups 0..194 are full (no checks)

    for (int g = 0; g < NG - 1; ++g) {
        const int nb = n_start + g * 16;
        {   // prefetch next group's rows (global_prefetch_b8)
            int npf = nb + 16 + h * 8;
            if (npf < N_) __builtin_prefetch(M + (size_t)npf * D_, 0, 1);
        }
        process_group<false>(nb, n_end, r, h, M, mask, qv, C, Ms, L, bv, bi);
    }
    // tail group: 3125 - 195*16 = 5 valid memories
    process_group<true>(n_start + (NG - 1) * 16, n_end, r, h, M, mask, qv, C, Ms, L, bv, bi);

    // ---- write partials ----
    float* zp  = ws;
    float* mp  = ws + (size_t)NREC * 16 * 128;
    float* lp  = mp + NREC * 16;
    float* bvp = lp + NREC * 16;
    int*   bip = (int*)(bvp + NREC * 16);

    // C VGPR v: lanes 0-15 -> row v, lanes 16-31 -> row v+8; column = r
#pragma unroll
    for (int t = 0; t < 8; ++t)
#pragma unroll
        for (int v = 0; v < 8; ++v)
            zp[((size_t)rec * 16 + v + 8 * h) * 128 + t * 16 + r] = C[t][v];

    {   // merge argmax across halves (smaller index wins ties)
        float bv2 = __shfl_xor(bv, 16, 32);
        int   bi2 = __shfl_xor(bi, 16, 32);
        if (bv2 > bv || (bv2 == bv && bi2 < bi)) { bv = bv2; bi = bi2; }
    }
    if (h == 0) {
        mp [rec * 16 + r] = Ms;
        lp [rec * 16 + r] = L;
        bvp[rec * 16 + r] = bv;
        bip[rec * 16 + r] = bi;
    }
}

__global__ __launch_bounds__(128)
void hopfield_pass2(const float* __restrict__ ws, float* __restrict__ out) {
    const int row   = blockIdx.x;     // 0..2047
    const int d     = threadIdx.x;    // 0..127
    const int btile = row >> 4;
    const int rl    = row & 15;

    const float* zp  = ws;
    const float* mp  = ws + (size_t)NREC * 16 * 128;
    const float* lp  = mp + NREC * 16;
    const float* bvp = lp + NREC * 16;
    const int*   bip = (const int*)(bvp + NREC * 16);

    float Mg = -3.4e38f;
#pragma unroll
    for (int k = 0; k < NSPLIT; ++k) {
        const int rec = btile * NSPLIT + k;
        Mg = fmaxf(Mg, mp[rec * 16 + rl]);
    }
    float Ls = 0.f, zacc = 0.f;
#pragma unroll
    for (int k = 0; k < NSPLIT; ++k) {
        const int rec = btile * NSPLIT + k;
        const float w = __expf(mp[rec * 16 + rl] - Mg);
        Ls   += w * lp[rec * 16 + rl];
        zacc += w * zp[((size_t)rec * 16 + rl) * 128 + d];
    }
    out[(size_t)row * D_ + d] = zacc / Ls;

    if (d == 0) {
        float bv = -3.4e38f; int bi = 0;
#pragma unroll
        for (int k = 0; k < NSPLIT; ++k) {
            const int rec = btile * NSPLIT + k;
            const float v = bvp[rec * 16 + rl];
            const int   i = bip[rec * 16 + rl];
            if (v > bv || (v == bv && i < bi)) { bv = v; bi = i; }
        }
        out[(size_t)B_ * D_ + row] = (float)bi;   // indices < 2^24: exact in f32
    }
}

extern "C" void kernel_launch(void* const* d_in, const int* in_sizes, int n_in,
                              void* d_out, int out_size, void* d_ws, size_t ws_size,
                              hipStream_t stream) {
    const float*         q    = (const float*)d_in[0];
    const float*         M    = (const float*)d_in[1];
    const unsigned char* mask = (const unsigned char*)d_in[2];
    float* out = (float*)d_out;
    float* ws  = (float*)d_ws;   // needs ~17.3 MB

    dim3 g1(NBT, NSPLIT);        // 128 x 16 single-wave workgroups
    hipLaunchKernelGGL(hopfield_pass1, g1, dim3(32), 0, stream, q, M, mask, ws);
    hipLaunchKernelGGL(hopfield_pass2, dim3(B_), dim3(128), 0, stream, ws, out);
}